// DispersiveLoss_53850299957877
// MI455X (gfx1250) — compile-verified
//
#include <hip/hip_runtime.h>
#include <math.h>

typedef float v2f __attribute__((ext_vector_type(2)));
typedef float v8f __attribute__((ext_vector_type(8)));

#define NROWS 2048
#define DIM   12288
#define TILES (NROWS / 16)     // 128 row tiles
#define CBLKS (NROWS / 64)     // 32  column blocks (64 columns each)

#define WMMA_F32(a, b, c) \
    __builtin_amdgcn_wmma_f32_16x16x4_f32(false, (a), false, (b), (short)0, (c), false, false)

// ---------------------------------------------------------------- init
__global__ void dl_init_acc(double* acc) {
    if (threadIdx.x < 4) acc[threadIdx.x] = 0.0;
}

// ---------------------------------------------------------------- per-row ||x||^2 and sum of norms
__global__ void dl_row_stats(const float* __restrict__ x,
                             float* __restrict__ sq,
                             double* __restrict__ acc) {
    __shared__ float red[256];
    const int row = blockIdx.x;
    const float4* xr = (const float4*)(x + (size_t)row * DIM);
    float s = 0.f;
    for (int i = threadIdx.x; i < DIM / 4; i += 256) {
        float4 v = xr[i];
        s += v.x * v.x + v.y * v.y + v.z * v.z + v.w * v.w;
    }
    red[threadIdx.x] = s;
    __syncthreads();
    for (int off = 128; off > 0; off >>= 1) {
        if (threadIdx.x < off) red[threadIdx.x] += red[threadIdx.x + off];
        __syncthreads();
    }
    if (threadIdx.x == 0) {
        float t = red[0];
        sq[row] = t;
        atomicAdd(&acc[3], (double)sqrtf(t));   // feat-norm accumulator
    }
}

// ---------------------------------------------------------------- Gram 16x64 strip per wave, fp32 WMMA
// One wave32 per (16-row, 64-col) block: 1 shared A fragment, 4 B fragments,
// 4 accumulators per K step -> 5 loads / 4 WMMAs (6.4 flops per L2 byte).
__global__ void __launch_bounds__(32)
dl_gram_tile(const float* __restrict__ x,
             const float* __restrict__ sq,
             double* __restrict__ acc) {
    const int tr = blockIdx.y;              // 16-row tile index  (i block)
    const int tb = blockIdx.x;              // 64-col block index (j block)
    if (tb * 4 + 3 < tr) return;            // block fully below diagonal: uniform exit

    const int lane = threadIdx.x;           // 0..31
    const int m    = lane & 15;
    const int half = lane >> 4;

    // fp32 WMMA fragment layouts (16x4 A / 4x16 B): lanes 0-15 hold (row, K=0..1),
    // lanes 16-31 hold (row, K=2..3). For X*X^T both operands are rows of X, so
    // both use the same per-lane address pattern: row base + k + 2*half.
    const float* pa = x + (size_t)(tr * 16 + m) * DIM + 2 * half;
    const float* pb = x + (size_t)(tb * 64 + m) * DIM + 2 * half;

    v8f c0 = {0.f,0.f,0.f,0.f,0.f,0.f,0.f,0.f};
    v8f c1 = c0, c2 = c0, c3 = c0;

#pragma unroll 4
    for (int k = 0; k < DIM; k += 4) {
        v2f a  = *(const v2f*)(pa + k);
        v2f b0 = *(const v2f*)(pb + k);                 // col tile 0 (+0    rows)
        v2f b1 = *(const v2f*)(pb + 16 * DIM + k);      // col tile 1 (+16   rows)
        v2f b2 = *(const v2f*)(pb + 32 * DIM + k);      // col tile 2 (+32   rows)
        v2f b3 = *(const v2f*)(pb + 48 * DIM + k);      // col tile 3 (+48   rows)
        c0 = WMMA_F32(a, b0, c0);
        c1 = WMMA_F32(a, b1, c1);
        c2 = WMMA_F32(a, b2, c2);
        c3 = WMMA_F32(a, b3, c3);
    }

    // C/D layout: VGPR r -> (M = r + 8*half, N = lane & 15)
    const float invD = 1.0f / (float)DIM;
    float s1 = 0.f, s2 = 0.f, se = 0.f;

#pragma unroll
    for (int t = 0; t < 4; ++t) {
        const v8f c = (t == 0) ? c0 : (t == 1) ? c1 : (t == 2) ? c2 : c3;
        const int j = tb * 64 + t * 16 + m;
        const float sqj = sq[j];
#pragma unroll
        for (int r = 0; r < 8; ++r) {
            const int i = tr * 16 + r + 8 * half;
            if (j > i) {
                float d2  = fmaxf(sq[i] + sqj - 2.0f * c[r], 0.f);
                float pdn = d2 * invD;
                s1 += pdn;
                s2 += pdn * pdn;
                se += expf(-2.0f * pdn);          // 1/TAU = 2
            }
        }
    }

    // wave32 reduction
    for (int off = 16; off > 0; off >>= 1) {
        s1 += __shfl_xor(s1, off, 32);
        s2 += __shfl_xor(s2, off, 32);
        se += __shfl_xor(se, off, 32);
    }
    if (lane == 0) {
        atomicAdd(&acc[0], (double)s1);
        atomicAdd(&acc[1], (double)s2);
        atomicAdd(&acc[2], (double)se);
    }
}

// ---------------------------------------------------------------- finalize 4 scalars
__global__ void dl_finalize(const double* __restrict__ acc, float* __restrict__ out) {
    if (threadIdx.x == 0 && blockIdx.x == 0) {
        const double Np  = (double)NROWS * (NROWS - 1) / 2.0;  // 2,096,128 pairs
        const double s1  = acc[0];
        const double s2  = acc[1];
        const double se  = acc[2];
        const double ns  = acc[3];
        const double mean = s1 / Np;
        double var  = (s2 - s1 * s1 / Np) / (Np - 1.0);        // unbiased
        if (var < 0.0) var = 0.0;
        const double loss = -log(se) + log(Np);
        out[0] = (float)loss;
        out[1] = (float)(ns / (double)NROWS);
        out[2] = (float)mean;
        out[3] = (float)sqrt(var);
    }
}

// ---------------------------------------------------------------- launch
extern "C" void kernel_launch(void* const* d_in, const int* in_sizes, int n_in,
                              void* d_out, int out_size, void* d_ws, size_t ws_size,
                              hipStream_t stream) {
    const float* x   = (const float*)d_in[0];
    float*       out = (float*)d_out;
    double*      acc = (double*)d_ws;                 // 4 doubles
    float*       sq  = (float*)((char*)d_ws + 64);    // 2048 floats

    dl_init_acc<<<1, 32, 0, stream>>>(acc);
    dl_row_stats<<<NROWS, 256, 0, stream>>>(x, sq, acc);

    dim3 grid(CBLKS, TILES);                          // 32 x 128 blocks, 1 wave each
    dl_gram_tile<<<grid, 32, 0, stream>>>(x, sq, acc);

    dl_finalize<<<1, 1, 0, stream>>>(acc, out);
}